// Embedding_54649163874491
// MI455X (gfx1250) — compile-verified
//
#include <hip/hip_runtime.h>

// ---------------------------------------------------------------------------
// Fused SpikingJelly block: Linear(63->128, no bias) + BatchNorm1d(train stats)
// + MultiStepLIF (tau=2, hard reset, threshold 1) on MI455X (gfx1250, wave32).
//
// Shapes: x[T=4, B=64, N=1024, P=63] f32, W[D=128, P=63], gamma/beta[128].
// Memory-bound: 66MB x + 134MB out; recompute GEMM in pass2 instead of
// materializing h (x fits in 192MB L2). GEMM uses V_WMMA_F32_16X16X4_F32.
// ---------------------------------------------------------------------------

typedef float v2f __attribute__((ext_vector_type(2)));
typedef float v8f __attribute__((ext_vector_type(8)));

#define T_DIM   4
#define P_DIM   63
#define D_DIM   128
#define BN_ROWS 65536                 // B*N
#define M_TOT   (T_DIM * BN_ROWS)     // 262144 rows of h
#define KPAD    64
#define KSTRIDE 66                    // LDS row stride (floats), avoids bank conflicts
#define NSAMP   262144.0f

#define P1_BLOCKS 512
#define P1_TILES  (M_TOT / 16 / P1_BLOCKS)   // 32 M-tiles per block

// ws layout (floats): [0:128) scale, [128:256) shift, [256 + blk*256 : ...) partials
#define WS_PART_OFF 256

// ---------------------------------------------------------------------------
// Pass 1: WMMA GEMM over all M_TOT rows; per-channel sum / sumsq partials.
// Block = 256 threads = 8 waves; wave w owns D columns [16w, 16w+16).
// ---------------------------------------------------------------------------
__global__ __launch_bounds__(256) void lif_stats_kernel(
    const float* __restrict__ x, const float* __restrict__ W,
    float* __restrict__ ws_part)
{
    __shared__ float lx[16 * KSTRIDE];

    const int tid  = threadIdx.x;
    const int lane = tid & 31;
    const int wave = tid >> 5;
    const int half = lane >> 4;   // which 16-lane half
    const int lm   = lane & 15;   // m (A/C) or n (B/C) within tile
    const int d0   = wave * 16;

    // Resident B fragments: Bmat[k][n] = W[(d0+n)*P + k]; K padded 63->64 with 0.
    v2f bf[16];
#pragma unroll
    for (int s = 0; s < 16; ++s) {
        int k0 = 4 * s + 2 * half;
        v2f b;
        b.x = (k0     < P_DIM) ? W[(d0 + lm) * P_DIM + k0]     : 0.0f;
        b.y = (k0 + 1 < P_DIM) ? W[(d0 + lm) * P_DIM + k0 + 1] : 0.0f;
        bf[s] = b;
    }

    float cs = 0.0f, cq = 0.0f;   // per-lane column partials (column d0+lm)

    for (int it = 0; it < P1_TILES; ++it) {
        const int row0 = (blockIdx.x * P1_TILES + it) * 16;

        __syncthreads();
#pragma unroll
        for (int j = 0; j < 4; ++j) {          // 1024 logical elems / 256 thr
            int e = tid + 256 * j;
            int m = e >> 6, k = e & 63;
            lx[m * KSTRIDE + k] = (k < P_DIM) ? x[(row0 + m) * P_DIM + k] : 0.0f;
        }
        __syncthreads();

        v8f acc = {0.f, 0.f, 0.f, 0.f, 0.f, 0.f, 0.f, 0.f};
#pragma unroll
        for (int s = 0; s < 16; ++s) {
            int k0 = 4 * s + 2 * half;
            v2f a = *(const v2f*)&lx[lm * KSTRIDE + k0];   // ds_load_b64, 8B aligned
            acc = __builtin_amdgcn_wmma_f32_16x16x4_f32(
                false, a, false, bf[s], (short)0, acc, false, false);
        }
#pragma unroll
        for (int i = 0; i < 8; ++i) { float h = acc[i]; cs += h; cq += h * h; }
    }

    // Column d0+lm lives in lanes lm and lm+16: fold halves, then store partials.
    cs += __shfl_xor(cs, 16, 32);
    cq += __shfl_xor(cq, 16, 32);
    if (lane < 16) {
        float* p = ws_part + (size_t)blockIdx.x * 256;
        p[d0 + lm]       = cs;
        p[128 + d0 + lm] = cq;
    }
}

// ---------------------------------------------------------------------------
// Finalize: deterministic fixed-order fold of partials -> scale/shift.
// ---------------------------------------------------------------------------
__global__ __launch_bounds__(128) void lif_finalize_kernel(
    const float* __restrict__ gamma, const float* __restrict__ beta,
    const float* __restrict__ ws_part, float* __restrict__ ws_sc)
{
    const int d = threadIdx.x;          // 0..127
    float s = 0.0f, q = 0.0f;
    for (int b = 0; b < P1_BLOCKS; ++b) {
        s += ws_part[b * 256 + d];
        q += ws_part[b * 256 + 128 + d];
    }
    float mean = s / NSAMP;
    float var  = q / NSAMP - mean * mean;          // biased, as torch BN
    float rstd = rsqrtf(var + 1e-5f);
    float sc   = gamma[d] * rstd;
    ws_sc[d]         = sc;
    ws_sc[128 + d]   = beta[d] - mean * sc;
}

// ---------------------------------------------------------------------------
// Pass 2: recompute GEMM per (bn-tile, t), affine BN, LIF over T=4, store spikes.
// Block handles 16 bn-rows x 128 D x 4 t; wave w owns D-tile w.
// ---------------------------------------------------------------------------
__global__ __launch_bounds__(256) void lif_main_kernel(
    const float* __restrict__ x, const float* __restrict__ W,
    const float* __restrict__ ws_sc, float* __restrict__ out)
{
    __shared__ float lx[T_DIM * 16 * KSTRIDE];    // ~16.5 KB

    const int tid  = threadIdx.x;
    const int lane = tid & 31;
    const int wave = tid >> 5;
    const int half = lane >> 4;
    const int lm   = lane & 15;
    const int d0   = wave * 16;
    const int bn0  = blockIdx.x * 16;

    // Stage x tiles for all 4 timesteps (4096 logical elems / 256 threads).
#pragma unroll
    for (int j = 0; j < 16; ++j) {
        int e = tid + 256 * j;
        int t = e >> 10;
        int r = e & 1023;
        int m = r >> 6, k = r & 63;
        lx[t * (16 * KSTRIDE) + m * KSTRIDE + k] =
            (k < P_DIM) ? x[(t * BN_ROWS + bn0 + m) * P_DIM + k] : 0.0f;
    }

    // Resident B fragments (same layout as pass 1).
    v2f bf[16];
#pragma unroll
    for (int s = 0; s < 16; ++s) {
        int k0 = 4 * s + 2 * half;
        v2f b;
        b.x = (k0     < P_DIM) ? W[(d0 + lm) * P_DIM + k0]     : 0.0f;
        b.y = (k0 + 1 < P_DIM) ? W[(d0 + lm) * P_DIM + k0 + 1] : 0.0f;
        bf[s] = b;
    }
    const float sc = ws_sc[d0 + lm];         // per-column BN affine
    const float sh = ws_sc[128 + d0 + lm];

    __syncthreads();

    v8f acc[T_DIM];
#pragma unroll
    for (int t = 0; t < T_DIM; ++t) {
        v8f a8 = {0.f, 0.f, 0.f, 0.f, 0.f, 0.f, 0.f, 0.f};
        const float* lxt = &lx[t * (16 * KSTRIDE)];
#pragma unroll
        for (int s = 0; s < 16; ++s) {
            int k0 = 4 * s + 2 * half;
            v2f a = *(const v2f*)&lxt[lm * KSTRIDE + k0];
            a8 = __builtin_amdgcn_wmma_f32_16x16x4_f32(
                false, a, false, bf[s], (short)0, a8, false, false);
        }
        acc[t] = a8;
    }

    // LIF: v' = (v + h)/2 ; spike = H(v' - 1) ; hard reset to 0.
    float v[8] = {0.f, 0.f, 0.f, 0.f, 0.f, 0.f, 0.f, 0.f};
#pragma unroll
    for (int t = 0; t < T_DIM; ++t) {
#pragma unroll
        for (int i = 0; i < 8; ++i) {
            float h = acc[t][i] * sc + sh;
            float vn = 0.5f * (v[i] + h);
            float spike = (vn >= 1.0f) ? 1.0f : 0.0f;
            v[i] = vn * (1.0f - spike);
            int mrow = i + 8 * half;                        // C-layout row
            out[(t * BN_ROWS + bn0 + mrow) * D_DIM + d0 + lm] = spike;
        }
    }
}

// ---------------------------------------------------------------------------
extern "C" void kernel_launch(void* const* d_in, const int* in_sizes, int n_in,
                              void* d_out, int out_size, void* d_ws, size_t ws_size,
                              hipStream_t stream)
{
    const float* x     = (const float*)d_in[0];
    const float* W     = (const float*)d_in[1];
    const float* gamma = (const float*)d_in[2];
    const float* beta  = (const float*)d_in[3];
    // d_in[4] = pe (scalar, unused: pe=False branch)

    float* out     = (float*)d_out;
    float* ws_sc   = (float*)d_ws;                 // 256 floats: scale|shift
    float* ws_part = (float*)d_ws + WS_PART_OFF;   // P1_BLOCKS*256 floats (~512KB)

    lif_stats_kernel<<<P1_BLOCKS, 256, 0, stream>>>(x, W, ws_part);
    lif_finalize_kernel<<<1, 128, 0, stream>>>(gamma, beta, ws_part, ws_sc);
    lif_main_kernel<<<BN_ROWS / 16, 256, 0, stream>>>(x, W, ws_sc, out);
}